// DeChunkLayer_16192026705996
// MI455X (gfx1250) — compile-verified
//
#include <hip/hip_runtime.h>

// ---------------------------------------------------------------------------
// DeChunk forward for MI455X (gfx1250, wave32).
// h_t = (1-q_t) h_{t-1} + q_t * x_t along the sorted-boundary axis, with the
// segment-gather fused into the scan (h_t -> out rows [pos[t], pos[t+1])).
// Intra-16-step scan runs on the fp32 WMMA pipe (V_WMMA_F32_16X16X4_F32 x4).
// The triangular transition matrix M is built cooperatively in LDS (column
// walk), so A-operands come from ds_loads. Chunks entirely past the boundary
// count nb are skipped (their summary is exactly U=0, Aprod=1).
// ---------------------------------------------------------------------------

#if defined(__HIP_DEVICE_COMPILE__) && !__has_builtin(__builtin_amdgcn_wmma_f32_16x16x4_f32)
#error "gfx1250 f32 WMMA builtin not available on this toolchain"
#endif

typedef float v2f __attribute__((ext_vector_type(2)));
typedef float v8f __attribute__((ext_vector_type(8)));

#define B_   4
#define L_   4096
#define D_   1024
#define NCH  16        // chunks per row
#define CHT  256       // timesteps per chunk
#define NT   16        // 16-step WMMA tiles per chunk
#define MPAD 17        // padded row stride for M in LDS (bank-conflict-free)

// ---------------------------------------------------------------------------
// Kernel 1: per-row stable-partition bookkeeping.
//   q[b,rank]   = clip(prob[b,l,1]) for boundary tokens in rank order, 0 past nb
//   pos[b,rank] = token position l of that boundary, L past nb
//   nbArr[b]    = boundary count
// ---------------------------------------------------------------------------
__global__ void dechunk_prep(const float* __restrict__ prob,
                             const int*   __restrict__ mask,
                             float* __restrict__ q,
                             int*   __restrict__ pos,
                             int*   __restrict__ nbArr) {
  __shared__ int scan[1024];
  const int b = blockIdx.x;
  const int t = threadIdx.x;
  const int l0 = t * 4;
  int m[4], s[4];
  int run = 0;
#pragma unroll
  for (int i = 0; i < 4; ++i) {
    m[i] = (mask[b * L_ + l0 + i] != 0) ? 1 : 0;
    run += m[i];
    s[i] = run;
  }
  scan[t] = run;
  __syncthreads();
  for (int off = 1; off < 1024; off <<= 1) {
    int add = (t >= off) ? scan[t - off] : 0;
    __syncthreads();
    scan[t] += add;
    __syncthreads();
  }
  const int excl = scan[t] - run;
#pragma unroll
  for (int i = 0; i < 4; ++i) {
    const int incl = excl + s[i];
    if (m[i]) {
      float p = prob[(size_t)(b * L_ + l0 + i) * 2 + 1];
      p = fminf(fmaxf(p, 1e-4f), 1.0f - 1e-4f);
      q[b * L_ + incl - 1]   = p;
      pos[b * L_ + incl - 1] = l0 + i;
    }
  }
  const int nb = scan[1023];
  for (int tt = nb + t; tt < L_; tt += 1024) {
    q[b * L_ + tt]   = 0.0f;
    pos[b * L_ + tt] = L_;
  }
  if (t == 0) nbArr[b] = nb;
}

// ---------------------------------------------------------------------------
// Kernel 2: per-chunk decay product Aprod[b,c] = prod_{t in chunk} (1 - q_t).
// One thread per (b,chunk); chunks past nb have q=0 -> product 1.
// ---------------------------------------------------------------------------
__global__ void dechunk_aprod(const float* __restrict__ q,
                              float* __restrict__ Aprod) {
  const int idx = threadIdx.x;                 // 64 = B_*NCH threads
  if (idx >= B_ * NCH) return;
  const int b = idx / NCH, c = idx % NCH;
  const float* qp = q + b * L_ + c * CHT;
  float p = 1.0f;
  for (int t = 0; t < CHT; ++t) p *= (1.0f - qp[t]);
  Aprod[idx] = p;
}

// ---------------------------------------------------------------------------
// Kernel 3/5: WMMA chunk scan (MODE-specialized). Block = 4 waves; each wave
// owns a 16-wide d-tile, block owns one (b, chunk), 16 tiles of 16 steps.
// H(16x16) = M(16x16) * X(16x16) + cumA (x) carry, M lower-triangular.
// MODE 0: carry=0, emit chunk summary U.
// MODE 1: carry from interchunk scan; accumulator row t is written directly
//         to its output segment out[b, pos[t]..pos[t+1]-1, dtile].
// ---------------------------------------------------------------------------
template <int MODE>
__global__ void dechunk_scan(const float* __restrict__ hidden,
                             const float* __restrict__ q,
                             const int*   __restrict__ pos,
                             const int*   __restrict__ nbArr,
                             const float* __restrict__ carryIn,
                             float* __restrict__ out,
                             float* __restrict__ U) {
  __shared__ float q_sh[16], a_sh[16], cumA_sh[16];
  __shared__ float M_sh[16 * MPAD];           // M_sh[r*MPAD + c]
  __shared__ float carry_sh[4][16];
  __shared__ int   pos_sh[CHT + 1];
  const int b    = blockIdx.x;
  const int c    = blockIdx.y;
  const int wave = threadIdx.x >> 5;
  const int lane = threadIdx.x & 31;
  const int col  = lane & 15;         // N column (and A-matrix row)
  const int hi   = lane >> 4;         // lane half selects K pair / row half
  const int d0   = blockIdx.z * 64 + wave * 16;

  // Chunks entirely past nb: q=0 everywhere -> identity update, no outputs.
  const int nb = nbArr[b];
  if (c * CHT >= nb) {
    if (MODE == 0) {
      if (hi == 0) U[(size_t)(b * NCH + c) * D_ + d0 + col] = 0.0f;
    }
    return;
  }

  if (hi == 0)
    carry_sh[wave][col] =
        (MODE == 1) ? carryIn[(size_t)(b * NCH + c) * D_ + d0 + col] : 0.0f;
  if (MODE == 1) {
    for (int idx = threadIdx.x; idx <= CHT; idx += 128) {
      const int tt = c * CHT + idx;
      pos_sh[idx] = (tt < L_) ? pos[b * L_ + tt] : L_;
    }
  }
  // Zero M once: the strict upper triangle is never overwritten afterwards.
  for (int idx = threadIdx.x; idx < 16 * MPAD; idx += 128) M_sh[idx] = 0.0f;
  __syncthreads();

  for (int i = 0; i < NT; ++i) {
    const int t0 = c * CHT + i * 16;
    if (threadIdx.x < 16) {
      const float qv = q[b * L_ + t0 + threadIdx.x];
      q_sh[threadIdx.x] = qv;
      a_sh[threadIdx.x] = 1.0f - qv;
    }
    __syncthreads();
    // Cooperative triangular build: thread cc walks column cc down the rows:
    //   M[cc][cc] = q_cc ;  M[r][cc] = M[r-1][cc] * a_r  (r > cc).
    if (threadIdx.x < 16) {
      const int cc = threadIdx.x;
      float v = q_sh[cc];
      M_sh[cc * MPAD + cc] = v;
      for (int r = cc + 1; r < 16; ++r) {
        v *= a_sh[r];
        M_sh[r * MPAD + cc] = v;
      }
    } else if (threadIdx.x == 16) {
      float cp = 1.0f;
      for (int tt = 0; tt < 16; ++tt) { cp *= a_sh[tt]; cumA_sh[tt] = cp; }
    }
    __syncthreads();

    // C init folds the incoming carry: C[row][n] = cumA[row]*carry[n].
    const float cv = carry_sh[wave][col];
    v8f acc;
#pragma unroll
    for (int vr = 0; vr < 8; ++vr) acc[vr] = cumA_sh[vr + 8 * hi] * cv;

    // A operand rows come straight from LDS (address-computed ds_loads).
    const float* mr = &M_sh[col * MPAD + 2 * hi];
    const float* xb = hidden + ((size_t)b * L_ + t0) * D_ + d0 + col;
#pragma unroll
    for (int k = 0; k < 4; ++k) {
      v2f A, Bv;
      A.x = mr[4 * k + 0];
      A.y = mr[4 * k + 1];
      const int rb = 4 * k + 2 * hi;
      Bv.x = xb[(size_t)rb * D_];
      Bv.y = xb[(size_t)(rb + 1) * D_];
      acc = __builtin_amdgcn_wmma_f32_16x16x4_f32(
          false, A, false, Bv, (short)0, acc, false, false);
    }

    if (MODE == 1) {
      // Fused segment scatter: row t feeds out rows [pos[t], pos[t+1]).
#pragma unroll
      for (int vr = 0; vr < 8; ++vr) {
        const int tloc = i * 16 + vr + 8 * hi;
        const int ls = pos_sh[tloc];
        const int le = pos_sh[tloc + 1];
        const float v = acc[vr];
        float* op = out + ((size_t)b * L_ + ls) * D_ + d0 + col;
#pragma clang loop unroll(disable)
        for (int l = ls; l < le; ++l) { *op = v; op += D_; }
      }
    }
    if (hi == 1) carry_sh[wave][col] = acc[7];  // h at row 15 -> next carry
    __syncthreads();
  }

  if (MODE == 0) {
    if (hi == 0)
      U[(size_t)(b * NCH + c) * D_ + d0 + col] = carry_sh[wave][col];
  }
}

// ---------------------------------------------------------------------------
// Kernel 4: inter-chunk scan (16 chunks) per (b,d) channel.
// ---------------------------------------------------------------------------
__global__ void dechunk_interchunk(const float* __restrict__ U,
                                   const float* __restrict__ Aprod,
                                   float* __restrict__ carryIn) {
  const int b = blockIdx.x;
  const int d = blockIdx.y * 256 + threadIdx.x;
  float carry = 0.0f;
  for (int c = 0; c < NCH; ++c) {
    carryIn[(size_t)(b * NCH + c) * D_ + d] = carry;
    carry = Aprod[b * NCH + c] * carry + U[(size_t)(b * NCH + c) * D_ + d];
  }
}

// ---------------------------------------------------------------------------
extern "C" void kernel_launch(void* const* d_in, const int* in_sizes, int n_in,
                              void* d_out, int out_size, void* d_ws, size_t ws_size,
                              hipStream_t stream) {
  const float* hidden = (const float*)d_in[0];   // (B,L,D) f32
  const float* prob   = (const float*)d_in[1];   // (B,L,2) f32
  const int*   mask   = (const int*)d_in[2];     // (B,L)   bool->int
  float* out = (float*)d_out;

  // Workspace layout (~0.7 MB total)
  char* w = (char*)d_ws;
  float* q       = (float*)w; w += (size_t)B_ * L_ * sizeof(float);
  int*   pos     = (int*)w;   w += (size_t)B_ * L_ * sizeof(int);
  float* U       = (float*)w; w += (size_t)B_ * NCH * D_ * sizeof(float);
  float* carryIn = (float*)w; w += (size_t)B_ * NCH * D_ * sizeof(float);
  float* Aprod   = (float*)w; w += (size_t)B_ * NCH * sizeof(float);
  int*   nbArr   = (int*)w;   w += (size_t)B_ * sizeof(int);

  dechunk_prep<<<B_, 1024, 0, stream>>>(prob, mask, q, pos, nbArr);

  dechunk_aprod<<<1, 64, 0, stream>>>(q, Aprod);

  dechunk_scan<0><<<dim3(B_, NCH, D_ / 64), 128, 0, stream>>>(
      hidden, q, pos, nbArr, nullptr, nullptr, U);

  dechunk_interchunk<<<dim3(B_, D_ / 256), 256, 0, stream>>>(U, Aprod, carryIn);

  dechunk_scan<1><<<dim3(B_, NCH, D_ / 64), 128, 0, stream>>>(
      hidden, q, pos, nbArr, carryIn, out, nullptr);
}